// NuclideGNN_41334765257112
// MI455X (gfx1250) — compile-verified
//
#include <hip/hip_runtime.h>

#define N_NODES 100000
#define NUM_E   3200000
#define FEAT    7
#define HID     128
#define EMB     64
#define EPS_BN  1e-5f

typedef __attribute__((ext_vector_type(2))) float v2f;
typedef __attribute__((ext_vector_type(8))) float v8f;

// ---------------- utility ----------------
__global__ void fill_kernel(float* __restrict__ p, float val, long count) {
  long i = (long)blockIdx.x * blockDim.x + threadIdx.x;
  long stride = (long)gridDim.x * blockDim.x;
  for (; i < count; i += stride) p[i] = val;
}

// deg[d] += 1 per edge (deg pre-filled with 1.0 for the self loop)
__global__ void degree_kernel(const int* __restrict__ dst, float* __restrict__ deg) {
  int e = blockIdx.x * blockDim.x + threadIdx.x;
  if (e < NUM_E) atomicAdd(&deg[dst[e]], 1.0f);
}

__global__ void rsqrt_kernel(const float* __restrict__ deg, float* __restrict__ dinv) {
  int i = blockIdx.x * blockDim.x + threadIdx.x;
  if (i < N_NODES) dinv[i] = rsqrtf(deg[i]);
}

// h = relu(x @ W_in + b_in), x: [N,7], W_in: [7,128]
__global__ void in_proj_kernel(const float* __restrict__ x, const float* __restrict__ Win,
                               const float* __restrict__ bin, float* __restrict__ h) {
  long n = blockIdx.x;
  int  c = threadIdx.x;
  float acc = bin[c];
#pragma unroll
  for (int f = 0; f < FEAT; ++f) acc = fmaf(x[n * FEAT + f], Win[f * HID + c], acc);
  h[n * HID + c] = fmaxf(acc, 0.0f);
}

// ---------------- fp32 WMMA GEMM: out[N x DO] = A[N x 128] @ W[128 x DO] ----------------
// One wave per 16-row strip; A tile staged in LDS (row pad 132 -> bank-conflict-free
// ds_load_b64 fragment reads); K swept 4 at a time with V_WMMA_F32_16X16X4_F32.
template <int DO>
__global__ void __launch_bounds__(32) gemm_wmma_kernel(const float* __restrict__ A,
                                                       const float* __restrict__ W,
                                                       float* __restrict__ out) {
  constexpr int K = 128;
  constexpr int LDSROW = 132;  // 128 + 4 pad: lane m hits bank (4m + k) % 64 -> all distinct
  __shared__ float sA[16 * LDSROW];
  const int  lane = threadIdx.x;
  const long row0 = (long)blockIdx.x * 16;

  // cooperative load: each iteration moves one 512B row, fully coalesced
#pragma unroll
  for (int i = 0; i < 16; ++i)
    ((float4*)(sA + i * LDSROW))[lane] = ((const float4*)(A + (row0 + i) * K))[lane];
  __syncthreads();

  const int m  = lane & 15;   // row (A) / col (B,C) within tile
  const int kh = lane >> 4;   // K-half select per ISA 16x4 layout

  v8f acc[DO / 16] = {};

  for (int k0 = 0; k0 < K; k0 += 4) {
    // A frag: lane m holds A[m][k0+2kh], A[m][k0+2kh+1]
    v2f a = *(const v2f*)(sA + m * LDSROW + k0 + kh * 2);
#pragma unroll
    for (int t = 0; t < DO / 16; ++t) {
      const int n = t * 16 + m;
      v2f b;
      b.x = W[(k0 + kh * 2 + 0) * DO + n];
      b.y = W[(k0 + kh * 2 + 1) * DO + n];
      acc[t] = __builtin_amdgcn_wmma_f32_16x16x4_f32(
          /*neg_a=*/false, a, /*neg_b=*/false, b,
          /*c_mod=*/(short)0, acc[t], /*reuse_a=*/false, /*reuse_b=*/false);
    }
  }

  // C/D layout: VGPR r, lanes 0-15 -> M=r, lanes 16-31 -> M=r+8
#pragma unroll
  for (int t = 0; t < DO / 16; ++t)
#pragma unroll
    for (int r = 0; r < 8; ++r)
      out[(row0 + r + kh * 8) * DO + t * 16 + m] = acc[t][r];
}

// ---------------- edge scatter: agg[dst] += hW[src] * dinv[src]*dinv[dst] ----------------
// DO=128: one wave per edge, each lane owns a float4 slice -> coalesced 512B gather.
template <int DO>
__global__ void scatter_kernel(const float* __restrict__ hW, const int* __restrict__ src,
                               const int* __restrict__ dst, const float* __restrict__ dinv,
                               float* __restrict__ agg) {
  constexpr int GPE = DO / 4;  // float4 groups per edge
  long gid = (long)blockIdx.x * blockDim.x + threadIdx.x;
  int  e   = (int)(gid / GPE);
  int  j   = (int)(gid % GPE);
  if (e >= NUM_E) return;
  const int s = src[e], d = dst[e];
  const float coef = dinv[s] * dinv[d];
  const float4 v = ((const float4*)(hW + (long)s * DO))[j];
  float* ap = agg + (long)d * DO + j * 4;
  atomicAdd(ap + 0, v.x * coef);
  atomicAdd(ap + 1, v.y * coef);
  atomicAdd(ap + 2, v.z * coef);
  atomicAdd(ap + 3, v.w * coef);
}

// ---------------- fused epilogue: self-loop + bias + BN(eval) [+ relu + residual] ----------------
template <int DO, bool RELU_RES>
__global__ void combine_kernel(const float* __restrict__ agg, const float* __restrict__ hW,
                               const float* __restrict__ hprev, const float* __restrict__ dinv,
                               const float* __restrict__ b, const float* __restrict__ g,
                               const float* __restrict__ be, const float* __restrict__ bm,
                               const float* __restrict__ bv, float* __restrict__ out) {
  long n = blockIdx.x;
  int  c = threadIdx.x;
  const float di  = dinv[n];
  float val = agg[n * DO + c] + hW[n * DO + c] * (di * di) + b[c];
  val = (val - bm[c]) * (g[c] * rsqrtf(bv[c] + EPS_BN)) + be[c];
  if (RELU_RES) val = fmaxf(val, 0.0f) + hprev[n * DO + c];
  out[n * DO + c] = val;
}

// ---------------- launcher ----------------
extern "C" void kernel_launch(void* const* d_in, const int* in_sizes, int n_in,
                              void* d_out, int out_size, void* d_ws, size_t ws_size,
                              hipStream_t stream) {
  const float* x   = (const float*)d_in[0];
  const int*   ei  = (const int*)d_in[1];
  const float* Win = (const float*)d_in[2];
  const float* bin = (const float*)d_in[3];
  const float* W0 = (const float*)d_in[4];  const float* b0 = (const float*)d_in[5];
  const float* g0 = (const float*)d_in[6];  const float* be0 = (const float*)d_in[7];
  const float* m0 = (const float*)d_in[8];  const float* v0 = (const float*)d_in[9];
  const float* W1 = (const float*)d_in[10]; const float* b1 = (const float*)d_in[11];
  const float* g1 = (const float*)d_in[12]; const float* be1 = (const float*)d_in[13];
  const float* m1 = (const float*)d_in[14]; const float* v1 = (const float*)d_in[15];
  const float* W2 = (const float*)d_in[16]; const float* b2 = (const float*)d_in[17];
  const float* g2 = (const float*)d_in[18]; const float* be2 = (const float*)d_in[19];
  const float* m2 = (const float*)d_in[20]; const float* v2 = (const float*)d_in[21];

  const int* src = ei;           // edge_index[0]
  const int* dst = ei + NUM_E;   // edge_index[1]

  // workspace layout (floats); all offsets 16B-aligned
  float* ws   = (float*)d_ws;
  float* deg  = ws;                              // N
  float* dinv = deg + N_NODES;                   // N
  float* hA   = dinv + N_NODES;                  // N*128
  float* hB   = hA + (long)N_NODES * HID;        // N*128
  float* hW   = hB + (long)N_NODES * HID;        // N*128
  float* agg  = hW + (long)N_NODES * HID;        // N*128

  const int MT = N_NODES / 16;  // 6250 exact strips

  // degrees (self-loop included) and D^-1/2
  fill_kernel<<<1024, 256, 0, stream>>>(deg, 1.0f, (long)N_NODES);
  degree_kernel<<<(NUM_E + 255) / 256, 256, 0, stream>>>(dst, deg);
  rsqrt_kernel<<<(N_NODES + 255) / 256, 256, 0, stream>>>(deg, dinv);

  // input projection
  in_proj_kernel<<<N_NODES, HID, 0, stream>>>(x, Win, bin, hA);

  // ----- layer 0: 128 -> 128, relu + residual(hA) -----
  gemm_wmma_kernel<128><<<MT, 32, 0, stream>>>(hA, W0, hW);
  fill_kernel<<<2048, 256, 0, stream>>>(agg, 0.0f, (long)N_NODES * 128);
  scatter_kernel<128><<<(NUM_E * 32 + 255) / 256, 256, 0, stream>>>(hW, src, dst, dinv, agg);
  combine_kernel<128, true><<<N_NODES, 128, 0, stream>>>(agg, hW, hA, dinv, b0, g0, be0, m0, v0, hB);

  // ----- layer 1: 128 -> 128, relu + residual(hB) -----
  gemm_wmma_kernel<128><<<MT, 32, 0, stream>>>(hB, W1, hW);
  fill_kernel<<<2048, 256, 0, stream>>>(agg, 0.0f, (long)N_NODES * 128);
  scatter_kernel<128><<<(NUM_E * 32 + 255) / 256, 256, 0, stream>>>(hW, src, dst, dinv, agg);
  combine_kernel<128, true><<<N_NODES, 128, 0, stream>>>(agg, hW, hB, dinv, b1, g1, be1, m1, v1, hA);

  // ----- layer 2: 128 -> 64, BN only -> d_out -----
  gemm_wmma_kernel<64><<<MT, 32, 0, stream>>>(hA, W2, hW);
  fill_kernel<<<2048, 256, 0, stream>>>(agg, 0.0f, (long)N_NODES * 64);
  scatter_kernel<64><<<(NUM_E * 16 + 255) / 256, 256, 0, stream>>>(hW, src, dst, dinv, agg);
  combine_kernel<64, false><<<N_NODES, 64, 0, stream>>>(agg, hW, nullptr, dinv, b2, g2, be2, m2, v2,
                                                        (float*)d_out);
}